// CNN_23742579212934
// MI455X (gfx1250) — compile-verified
//
#include <hip/hip_runtime.h>

#define BN_EPS 1e-5f

typedef __attribute__((ext_vector_type(2))) float v2f;
typedef __attribute__((ext_vector_type(8))) float v8f;

// ---------------------------------------------------------------------------
// K1: sparse conv1 (C 1->8) + per-block BN partial sums (deterministic)
// ---------------------------------------------------------------------------
__global__ __launch_bounds__(256)
void conv1_kernel(const float* __restrict__ f0, const int* __restrict__ nbr0,
                  const float* __restrict__ W1, const float* __restrict__ b1,
                  float* __restrict__ y1, float* __restrict__ part1, int n0) {
    __shared__ float w1s[72];
    __shared__ float b1s[8];
    __shared__ float red[8][16];
    const int tid = threadIdx.x;
    if (tid < 72) w1s[tid] = W1[tid];
    if (tid < 8)  b1s[tid] = b1[tid];
    __syncthreads();

    const int i = blockIdx.x * 256 + tid;
    float acc[8];
#pragma unroll
    for (int d = 0; d < 8; ++d) acc[d] = 0.f;

    if (i < n0) {
#pragma unroll
        for (int d = 0; d < 8; ++d) acc[d] = b1s[d];
#pragma unroll
        for (int k = 0; k < 9; ++k) {
            const int idx = nbr0[i * 9 + k];
            const float g = (idx >= 0) ? f0[idx] : 0.f;
#pragma unroll
            for (int d = 0; d < 8; ++d) acc[d] = fmaf(g, w1s[k * 8 + d], acc[d]);
        }
        float4* dst = (float4*)(y1 + (size_t)i * 8);
        dst[0] = make_float4(acc[0], acc[1], acc[2], acc[3]);
        dst[1] = make_float4(acc[4], acc[5], acc[6], acc[7]);
    }

    // block partial sums for BN1 (8 sums + 8 sumsqs), wave32 shuffle tree
    float s[8], ss[8];
#pragma unroll
    for (int d = 0; d < 8; ++d) { s[d] = acc[d]; ss[d] = acc[d] * acc[d]; }
#pragma unroll
    for (int off = 16; off > 0; off >>= 1) {
#pragma unroll
        for (int d = 0; d < 8; ++d) {
            s[d]  += __shfl_down(s[d],  off, 32);
            ss[d] += __shfl_down(ss[d], off, 32);
        }
    }
    const int wv = tid >> 5, ln = tid & 31;
    if (ln == 0) {
#pragma unroll
        for (int d = 0; d < 8; ++d) { red[wv][d] = s[d]; red[wv][8 + d] = ss[d]; }
    }
    __syncthreads();
    if (tid < 16) {
        float p = 0.f;
#pragma unroll
        for (int w = 0; w < 8; ++w) p += red[w][tid];
        part1[blockIdx.x * 16 + tid] = p;
    }
}

// ---------------------------------------------------------------------------
// K2/K6: fold BN partials -> per-channel affine: a = g*rsqrt(var+eps), b = beta - mu*a
// part layout: [nblk][2C]  (C sums, then C sumsqs).  stats: [a[0..C), b[0..C)]
// ---------------------------------------------------------------------------
__global__ __launch_bounds__(64)
void bn_finalize_kernel(const float* __restrict__ part, int nblk, int C, float invN,
                        const float* __restrict__ gamma, const float* __restrict__ beta,
                        float* __restrict__ stats) {
    __shared__ float sm[64];
    const int t = threadIdx.x;
    if (t < 2 * C) {
        float acc = 0.f;
        for (int b = 0; b < nblk; ++b) acc += part[b * 2 * C + t];
        sm[t] = acc;
    }
    __syncthreads();
    if (t < C) {
        const float mu  = sm[t] * invN;
        const float var = sm[C + t] * invN - mu * mu;
        const float a   = gamma[t] * rsqrtf(var + BN_EPS);
        stats[t]     = a;
        stats[C + t] = beta[t] - mu * a;
    }
}

// ---------------------------------------------------------------------------
// K3: pool1 = max over children of relu(affine(y1))   (8 channels)
// ---------------------------------------------------------------------------
__global__ __launch_bounds__(256)
void pool1_kernel(const float* __restrict__ y1, const int* __restrict__ pmap1,
                  const float* __restrict__ st1, float* __restrict__ x1p, int n1) {
    __shared__ float a[8], bb[8];
    if (threadIdx.x < 8)  a[threadIdx.x]  = st1[threadIdx.x];
    if (threadIdx.x < 8)  bb[threadIdx.x] = st1[8 + threadIdx.x];
    __syncthreads();
    const int o = blockIdx.x * 256 + threadIdx.x;
    if (o >= n1) return;
    float m[8];
#pragma unroll
    for (int d = 0; d < 8; ++d) m[d] = -1e30f;
#pragma unroll
    for (int j = 0; j < 4; ++j) {
        const int idx = pmap1[o * 4 + j];
        if (idx >= 0) {
            const float4* src = (const float4*)(y1 + (size_t)idx * 8);
            const float4 lo = src[0], hi = src[1];
            const float v[8] = { lo.x, lo.y, lo.z, lo.w, hi.x, hi.y, hi.z, hi.w };
#pragma unroll
            for (int d = 0; d < 8; ++d) {
                const float t = fmaxf(fmaf(v[d], a[d], bb[d]), 0.f);
                m[d] = fmaxf(m[d], t);
            }
        }
    }
    float4* dst = (float4*)(x1p + (size_t)o * 8);
    dst[0] = make_float4(m[0], m[1], m[2], m[3]);
    dst[1] = make_float4(m[4], m[5], m[6], m[7]);
}

// ---------------------------------------------------------------------------
// K4: sparse conv2 (C 8->16) as WMMA GEMM:  y2[16-row tile] = A(16x72) * B(72x16) + b2
// One wave per 16-row tile, 8 waves (128 rows) per block.
// A panel in LDS, row stride 74 floats (bank-conflict free, float2 aligned).
// ---------------------------------------------------------------------------
#define A_STRIDE 74
#define A_WAVE   (16 * A_STRIDE)

__global__ __launch_bounds__(256)
void conv2_wmma_kernel(const float* __restrict__ x1p, const int* __restrict__ nbr1,
                       const float* __restrict__ W2, const float* __restrict__ b2,
                       float* __restrict__ y2, int n1) {
    __shared__ float w2s[72 * 16];   // B panel: [K=72][N=16]
    __shared__ float b2s[16];
    __shared__ float As[8 * A_WAVE]; // per-wave A panels: [16 rows][74]

    const int tid  = threadIdx.x;
    const int wave = tid >> 5;
    const int lane = tid & 31;

    // stage W2 (global layout [9][8][16] == [72][16]) and bias
    for (int t = tid; t < 72 * 16; t += 256) w2s[t] = W2[t];
    if (tid < 16) b2s[tid] = b2[tid];

    const int rowbase = blockIdx.x * 128 + wave * 16;

    // stage A panel: 16 rows x 9 neighbors x 8 channels, zeros for missing
    if (rowbase < n1) {
        for (int t = lane; t < 144; t += 32) {
            const int r = t / 9;
            const int k = t - 9 * r;
            float4 lo = make_float4(0.f, 0.f, 0.f, 0.f);
            float4 hi = make_float4(0.f, 0.f, 0.f, 0.f);
            const int gr = rowbase + r;
            if (gr < n1) {
                const int idx = nbr1[gr * 9 + k];
                if (idx >= 0) {
                    const float4* src = (const float4*)(x1p + (size_t)idx * 8);
                    lo = src[0]; hi = src[1];
                }
            }
            float2* dst = (float2*)&As[wave * A_WAVE + r * A_STRIDE + k * 8];
            dst[0] = make_float2(lo.x, lo.y);
            dst[1] = make_float2(lo.z, lo.w);
            dst[2] = make_float2(hi.x, hi.y);
            dst[3] = make_float2(hi.z, hi.w);
        }
    }
    __syncthreads();

    if (rowbase >= n1) return;            // wave-uniform: EXEC stays all-ones

    const int grp = lane >> 4;            // half-wave group
    const int col = lane & 15;            // N index (and A's M index)

    v8f c;
#pragma unroll
    for (int v = 0; v < 8; ++v) c[v] = b2s[col];   // fold bias into C

    const float* Ap = &As[wave * A_WAVE + col * A_STRIDE + 2 * grp];
    const float* Bp = &w2s[(2 * grp) * 16 + col];

#pragma unroll
    for (int s = 0; s < 18; ++s) {        // K = 72 = 18 * 4
        v2f a = *(const v2f*)(Ap + 4 * s);            // K = 4s+2g, 4s+2g+1
        v2f b;
        b.x = Bp[(4 * s) * 16];                       // K = 4s+2g
        b.y = Bp[(4 * s) * 16 + 16];                  // K = 4s+2g+1
        c = __builtin_amdgcn_wmma_f32_16x16x4_f32(false, a, false, b,
                                                  (short)0, c, false, false);
    }

    // C/D layout: vgpr v, lanes 0-15 -> M=v ; lanes 16-31 -> M=v+8 ; N=lane&15
#pragma unroll
    for (int v = 0; v < 8; ++v) {
        const int r = rowbase + v + 8 * grp;
        if (r < n1) y2[(size_t)r * 16 + col] = c[v];
    }
}

// ---------------------------------------------------------------------------
// K5: deterministic grid-stride reduction of y2 -> BN2 partials [256][32]
// ---------------------------------------------------------------------------
__global__ __launch_bounds__(256)
void reduce_y2_kernel(const float* __restrict__ y2, int n1, float* __restrict__ part2) {
    __shared__ float red[8][32];
    float s[16], ss[16];
#pragma unroll
    for (int d = 0; d < 16; ++d) { s[d] = 0.f; ss[d] = 0.f; }

    const int stride = gridDim.x * 256;
    for (int r = blockIdx.x * 256 + threadIdx.x; r < n1; r += stride) {
        const float4* row = (const float4*)(y2 + (size_t)r * 16);
#pragma unroll
        for (int q = 0; q < 4; ++q) {
            const float4 v = row[q];
            const float e[4] = { v.x, v.y, v.z, v.w };
#pragma unroll
            for (int j = 0; j < 4; ++j) {
                const int d = q * 4 + j;
                s[d]  += e[j];
                ss[d] += e[j] * e[j];
            }
        }
    }
#pragma unroll
    for (int off = 16; off > 0; off >>= 1) {
#pragma unroll
        for (int d = 0; d < 16; ++d) {
            s[d]  += __shfl_down(s[d],  off, 32);
            ss[d] += __shfl_down(ss[d], off, 32);
        }
    }
    const int wv = threadIdx.x >> 5, ln = threadIdx.x & 31;
    if (ln == 0) {
#pragma unroll
        for (int d = 0; d < 16; ++d) { red[wv][d] = s[d]; red[wv][16 + d] = ss[d]; }
    }
    __syncthreads();
    if (threadIdx.x < 32) {
        float p = 0.f;
#pragma unroll
        for (int w = 0; w < 8; ++w) p += red[w][threadIdx.x];
        part2[blockIdx.x * 32 + threadIdx.x] = p;
    }
}

// ---------------------------------------------------------------------------
// K7: pool2 (16 ch) + FC(16->2) fused
// ---------------------------------------------------------------------------
__global__ __launch_bounds__(256)
void pool2_fc_kernel(const float* __restrict__ y2, const int* __restrict__ pmap2,
                     const float* __restrict__ st2, const float* __restrict__ Wfc,
                     const float* __restrict__ bfc, float* __restrict__ out, int n2) {
    __shared__ float a[16], bb[16], wfc[32], bf[2];
    if (threadIdx.x < 16) { a[threadIdx.x] = st2[threadIdx.x]; bb[threadIdx.x] = st2[16 + threadIdx.x]; }
    if (threadIdx.x < 32) wfc[threadIdx.x] = Wfc[threadIdx.x];
    if (threadIdx.x < 2)  bf[threadIdx.x]  = bfc[threadIdx.x];
    __syncthreads();

    const int o = blockIdx.x * 256 + threadIdx.x;
    if (o >= n2) return;

    float m[16];
#pragma unroll
    for (int d = 0; d < 16; ++d) m[d] = -1e30f;
#pragma unroll
    for (int j = 0; j < 4; ++j) {
        const int idx = pmap2[o * 4 + j];
        if (idx >= 0) {
            const float4* row = (const float4*)(y2 + (size_t)idx * 16);
#pragma unroll
            for (int q = 0; q < 4; ++q) {
                const float4 v = row[q];
                const float e[4] = { v.x, v.y, v.z, v.w };
#pragma unroll
                for (int k = 0; k < 4; ++k) {
                    const int d = q * 4 + k;
                    const float t = fmaxf(fmaf(e[k], a[d], bb[d]), 0.f);
                    m[d] = fmaxf(m[d], t);
                }
            }
        }
    }
    float o0 = bf[0], o1 = bf[1];
#pragma unroll
    for (int d = 0; d < 16; ++d) {
        o0 = fmaf(m[d], wfc[d * 2 + 0], o0);
        o1 = fmaf(m[d], wfc[d * 2 + 1], o1);
    }
    out[(size_t)o * 2 + 0] = o0;
    out[(size_t)o * 2 + 1] = o1;
}

// ---------------------------------------------------------------------------
extern "C" void kernel_launch(void* const* d_in, const int* in_sizes, int n_in,
                              void* d_out, int out_size, void* d_ws, size_t ws_size,
                              hipStream_t stream) {
    const float* f0   = (const float*)d_in[0];
    const float* W1   = (const float*)d_in[1];
    const float* b1   = (const float*)d_in[2];
    const float* g1   = (const float*)d_in[3];
    const float* be1  = (const float*)d_in[4];
    const float* W2   = (const float*)d_in[5];
    const float* b2   = (const float*)d_in[6];
    const float* g2   = (const float*)d_in[7];
    const float* be2  = (const float*)d_in[8];
    const float* Wfc  = (const float*)d_in[9];
    const float* bfc  = (const float*)d_in[10];
    const int* nbr0   = (const int*)d_in[11];
    const int* pmap1  = (const int*)d_in[12];
    const int* nbr1   = (const int*)d_in[13];
    const int* pmap2  = (const int*)d_in[14];

    const int n0 = in_sizes[0];
    const int n1 = in_sizes[12] / 4;
    const int n2 = in_sizes[14] / 4;
    (void)n_in; (void)out_size; (void)ws_size;

    const int nblk1 = (n0 + 255) / 256;
    const int RB    = 256;   // fixed reduce grid for determinism

    float* ws = (float*)d_ws;
    size_t off = 0;
    auto al = [](size_t x) { return (x + 255) & ~(size_t)255; };
    float* y1    = ws + off; off = al(off + (size_t)n0 * 8);
    float* x1p   = ws + off; off = al(off + (size_t)n1 * 8);
    float* y2    = ws + off; off = al(off + (size_t)n1 * 16);
    float* part1 = ws + off; off = al(off + (size_t)nblk1 * 16);
    float* part2 = ws + off; off = al(off + (size_t)RB * 32);
    float* st1   = ws + off; off = al(off + 16);
    float* st2   = ws + off; off = al(off + 32);

    conv1_kernel<<<nblk1, 256, 0, stream>>>(f0, nbr0, W1, b1, y1, part1, n0);
    bn_finalize_kernel<<<1, 64, 0, stream>>>(part1, nblk1, 8, 1.f / (float)n0, g1, be1, st1);
    pool1_kernel<<<(n1 + 255) / 256, 256, 0, stream>>>(y1, pmap1, st1, x1p, n1);
    conv2_wmma_kernel<<<(n1 + 127) / 128, 256, 0, stream>>>(x1p, nbr1, W2, b2, y2, n1);
    reduce_y2_kernel<<<RB, 256, 0, stream>>>(y2, n1, part2);
    bn_finalize_kernel<<<1, 64, 0, stream>>>(part2, RB, 16, 1.f / (float)n1, g2, be2, st2);
    pool2_fc_kernel<<<(n2 + 255) / 256, 256, 0, stream>>>(y2, pmap2, st2, Wfc, bfc,
                                                          (float*)d_out, n2);
}